// GCNEncoder_75024488726867
// MI455X (gfx1250) — compile-verified
//
#include <hip/hip_runtime.h>

#define HIDDEN 128
#define NCONV  3

typedef __attribute__((ext_vector_type(2))) float v2f;
typedef __attribute__((ext_vector_type(4))) float v4f;
typedef __attribute__((ext_vector_type(8))) float v8f;

// ---------------------------------------------------------------------------
// GEMM: X = A(N x 128) * W(128 x 128) using V_WMMA_F32_16X16X4_F32.
// One wave32 computes one 16x16 tile of X. K loop = 128/4 = 32 WMMAs.
//
// A-fragment layout (ISA 7.12.2, 32-bit A 16x4): lane l in [0,15] holds row l,
// VGPR0 = K0, VGPR1 = K1; lanes 16..31 hold K2/K3 of the same rows.
// B-fragment (4x16): lane l holds col l; VGPR0 = K0 (K2 for upper half),
// VGPR1 = K1 (K3). C/D: VGPR r holds row r (lanes 0-15) / row r+8 (16-31).
// ---------------------------------------------------------------------------
__global__ __launch_bounds__(128) void gcn_gemm_wmma(
    const float* __restrict__ A, const float* __restrict__ W,
    float* __restrict__ X, int N) {
  const int lane = threadIdx.x & 31;
  const int half = lane >> 4;       // 0: K0/K1, 1: K2/K3
  const int l    = lane & 15;
  const int mtile = blockIdx.x;
  const int ntile = blockIdx.y * blockDim.y + threadIdx.y;   // 0..7

  int row = mtile * 16 + l;
  if (row >= N) row = N - 1;                      // clamp (N%16==0 in practice)
  const float* arow = A + (size_t)row * HIDDEN + 2 * half;
  const int ncol    = ntile * 16 + l;
  const float* wcol = W + (size_t)(2 * half) * HIDDEN + ncol;

  v8f c = {0.f, 0.f, 0.f, 0.f, 0.f, 0.f, 0.f, 0.f};
  for (int k0 = 0; k0 < HIDDEN; k0 += 4) {
    v2f a, b;
    a.x = arow[k0];                               // A[row][k0 + 2*half]
    a.y = arow[k0 + 1];                           // A[row][k0 + 2*half + 1]
    b.x = wcol[(size_t)k0 * HIDDEN];              // W[k0 + 2*half][ncol]
    b.y = wcol[(size_t)(k0 + 1) * HIDDEN];        // W[k0 + 2*half + 1][ncol]
    c = __builtin_amdgcn_wmma_f32_16x16x4_f32(
        /*neg_a=*/false, a, /*neg_b=*/false, b,
        /*c_mod=*/(short)0, c, /*reuse_a=*/false, /*reuse_b=*/false);
  }

  const int orow = mtile * 16 + half * 8;
#pragma unroll
  for (int r = 0; r < 8; ++r) {
    if (orow + r < N) X[(size_t)(orow + r) * HIDDEN + ncol] = c[r];
  }
}

// acc[n][h] = bias[h]
__global__ __launch_bounds__(256) void gcn_init_bias(
    float* __restrict__ acc, const float* __restrict__ bias, int total) {
  int i = blockIdx.x * blockDim.x + threadIdx.x;
  if (i < total) acc[i] = bias[i & (HIDDEN - 1)];
}

// One wave per edge; lane owns 4 features. msg = relu(X[src] + edge_attr[e]);
// acc[dst] += msg via global_atomic_add_f32. edge_attr streamed non-temporally
// so the 307MB/conv stream does not evict X/acc from the 192MB L2.
__global__ __launch_bounds__(256) void gcn_edge_scatter(
    const float* __restrict__ X, const int* __restrict__ ei,
    const float* __restrict__ ea, float* __restrict__ acc, int E) {
  int t = blockIdx.x * blockDim.x + threadIdx.x;
  int e = t >> 5;
  if (e >= E) return;
  int lane = t & 31;

  int s = ei[e];          // src
  int d = ei[E + e];      // dst

  const v4f* xs = (const v4f*)(X + (size_t)s * HIDDEN);
  const v4f* as = (const v4f*)(ea + (size_t)e * HIDDEN);
  v4f x = xs[lane];
  v4f a = __builtin_nontemporal_load(&as[lane]);   // TH=NT stream

  v4f m;
  m.x = fmaxf(x.x + a.x, 0.f);
  m.y = fmaxf(x.y + a.y, 0.f);
  m.z = fmaxf(x.z + a.z, 0.f);
  m.w = fmaxf(x.w + a.w, 0.f);

  float* p = acc + (size_t)d * HIDDEN + lane * 4;
  __hip_atomic_fetch_add(p + 0, m.x, __ATOMIC_RELAXED, __HIP_MEMORY_SCOPE_AGENT);
  __hip_atomic_fetch_add(p + 1, m.y, __ATOMIC_RELAXED, __HIP_MEMORY_SCOPE_AGENT);
  __hip_atomic_fetch_add(p + 2, m.z, __ATOMIC_RELAXED, __HIP_MEMORY_SCOPE_AGENT);
  __hip_atomic_fetch_add(p + 3, m.w, __ATOMIC_RELAXED, __HIP_MEMORY_SCOPE_AGENT);
}

// out = (apply_relu ? relu(acc) : acc) + cur
__global__ __launch_bounds__(256) void gcn_epilogue(
    const float* __restrict__ acc, const float* __restrict__ cur,
    float* __restrict__ out, int apply_relu, int total) {
  int i = blockIdx.x * blockDim.x + threadIdx.x;
  if (i >= total) return;
  float h = acc[i];
  if (apply_relu) h = fmaxf(h, 0.f);
  out[i] = h + cur[i];
}

extern "C" void kernel_launch(void* const* d_in, const int* in_sizes, int n_in,
                              void* d_out, int out_size, void* d_ws, size_t ws_size,
                              hipStream_t stream) {
  const float* z  = (const float*)d_in[0];
  const int*   ei = (const int*)d_in[1];     // [2, E] flat: src then dst
  const float* ea = (const float*)d_in[2];
  const float* W  = (const float*)d_in[3];   // [3, 128, 128]
  const float* b  = (const float*)d_in[4];   // [3, 128]

  const int N = in_sizes[0] / HIDDEN;
  const int E = in_sizes[1] / 2;
  const int total = N * HIDDEN;

  float* ws   = (float*)d_ws;
  float* X    = ws;                       // GEMM output     [N,128]
  float* buf1 = ws + (size_t)total;       // hidden ping     [N,128]
  float* buf2 = ws + (size_t)2 * total;   // hidden pong     [N,128]
  float* out  = (float*)d_out;

  const dim3 gemm_blk(32, 4);
  const dim3 gemm_grd((N + 15) / 16, (HIDDEN / 16) / 4);
  const int eb = 256;
  const long ethreads = (long)E * 32;

  const float* cur = z;
  for (int i = 0; i < NCONV; ++i) {
    const float* Wi = W + (size_t)i * HIDDEN * HIDDEN;
    const float* bi = b + (size_t)i * HIDDEN;
    float* acc = (i == 0) ? buf1 : (i == 1) ? buf2 : buf1;
    float* dst = (i == NCONV - 1) ? out : acc;   // last conv writes d_out

    gcn_gemm_wmma<<<gemm_grd, gemm_blk, 0, stream>>>(cur, Wi, X, N);
    gcn_init_bias<<<(total + 255) / 256, 256, 0, stream>>>(acc, bi, total);
    gcn_edge_scatter<<<(unsigned)((ethreads + eb - 1) / eb), eb, 0, stream>>>(
        X, ei, ea, acc, E);
    gcn_epilogue<<<(total + 255) / 256, 256, 0, stream>>>(
        acc, cur, dst, (i < NCONV - 1) ? 1 : 0, total);
    cur = dst;
  }
}